// GCNEncoder_18408229830960
// MI455X (gfx1250) — compile-verified
//
#include <hip/hip_runtime.h>
#include <cstdint>
#include <cstddef>

typedef float v2f __attribute__((ext_vector_type(2)));
typedef float v8f __attribute__((ext_vector_type(8)));

// ---------------------------------------------------------------------------
// Native fire-and-forget f32 atomic add at device scope (L2 atomic units).
// Avoids any chance of a CAS-loop lowering from generic atomicAdd(float*).
// ---------------------------------------------------------------------------
__device__ __forceinline__ void atomic_add_f32_dev(float* p, float v) {
    asm volatile("global_atomic_add_f32 %0, %1, off scope:SCOPE_DEV"
                 :: "v"(p), "v"(v)
                 : "memory");
}

// ---------------------------------------------------------------------------
// Zero fill
// ---------------------------------------------------------------------------
__global__ void zero_kernel(float* __restrict__ p, size_t n) {
    size_t g = (size_t)blockIdx.x * blockDim.x + threadIdx.x;
    if (g < n) p[g] = 0.0f;
}

// ---------------------------------------------------------------------------
// Degree count over dst (A + I handled later by +1)
// ---------------------------------------------------------------------------
__global__ void deg_kernel(const int* __restrict__ dst, unsigned* __restrict__ deg, int E) {
    int e = blockIdx.x * blockDim.x + threadIdx.x;
    if (e < E) atomicAdd(&deg[dst[e]], 1u);
}

// deg (u32 counts) -> dinv (f32 rsqrt(deg+1)) in place
__global__ void dinv_kernel(float* __restrict__ degf, int N) {
    int i = blockIdx.x * blockDim.x + threadIdx.x;
    if (i < N) {
        unsigned c = ((const unsigned*)degf)[i];
        degf[i] = rsqrtf((float)c + 1.0f);
    }
}

// ---------------------------------------------------------------------------
// GEMM: out[M,NC] = A[M,64] @ W[64,NC] using V_WMMA_F32_16X16X4_F32.
// One wave computes one 16x16 tile; K looped in 16 steps of 4.
// f32 A 16x4 layout (ISA 7.12.2): lanes 0-15 hold M=lane, VGPR0=K+0,VGPR1=K+1;
// lanes 16-31 hold M=lane-16, VGPR0=K+2, VGPR1=K+3.  B 4x16 mirrors by rows.
// C/D: VGPR j holds row (j + 8*(lane>=16)), col = lane&15.
// Requires M % 16 == 0 (true here: 100000 = 16*6250).
// ---------------------------------------------------------------------------
template <int NC>
__global__ void gemm_wmma_kernel(const float* __restrict__ A,
                                 const float* __restrict__ W,
                                 float* __restrict__ out, int M) {
    constexpr int NT = NC / 16;
    int wave   = blockIdx.x * (blockDim.x >> 5) + (threadIdx.x >> 5);
    int mtiles = M >> 4;
    if (wave >= mtiles * NT) return;          // wave-uniform guard (EXEC all-ones below)
    int mt    = wave / NT;
    int nt    = wave - mt * NT;
    int lane  = threadIdx.x & 31;
    int l16   = lane & 15;
    int lhalf = lane >> 4;                    // 0: K+0/K+1, 1: K+2/K+3

    const float* arow = A + (size_t)(mt * 16 + l16) * 64 + 2 * lhalf;
    const float* bcol = W + (size_t)(2 * lhalf) * NC + nt * 16 + l16;

    v8f c = {0.f, 0.f, 0.f, 0.f, 0.f, 0.f, 0.f, 0.f};
#pragma unroll
    for (int k = 0; k < 64; k += 4) {
        v2f a = *(const v2f*)(arow + k);      // K = k+2*lhalf, k+2*lhalf+1
        v2f b;
        b.x = bcol[(size_t)k * NC];           // row K = k+2*lhalf
        b.y = bcol[(size_t)(k + 1) * NC];     // row K = k+1+2*lhalf
        c = __builtin_amdgcn_wmma_f32_16x16x4_f32(
                /*neg_a=*/false, a, /*neg_b=*/false, b,
                /*c_mod=*/(short)0, c, /*reuse_a=*/false, /*reuse_b=*/false);
    }

    float* orow = out + (size_t)(mt * 16 + 8 * lhalf) * NC + nt * 16 + l16;
#pragma unroll
    for (int j = 0; j < 8; ++j) orow[(size_t)j * NC] = c[j];
}

// ---------------------------------------------------------------------------
// Edge scatter: one 32-lane wave per edge, F/32 features per lane.
// Consecutive lanes hit consecutive features -> fully coalesced gather+atomic.
// ---------------------------------------------------------------------------
template <int F>
__global__ void scatter_kernel(const float* __restrict__ xw,
                               const int* __restrict__ src,
                               const int* __restrict__ dst,
                               const float* __restrict__ dinv,
                               float* __restrict__ agg, int E) {
    int e = blockIdx.x * (blockDim.x >> 5) + (threadIdx.x >> 5);
    if (e >= E) return;
    int lane = threadIdx.x & 31;
    int s = src[e];
    int d = dst[e];
    float coef = dinv[s] * dinv[d];
    const float* xs = xw + (size_t)s * F;
    float*       ad = agg + (size_t)d * F;
#pragma unroll
    for (int j = 0; j < F; j += 32)
        atomic_add_f32_dev(ad + j + lane, xs[j + lane] * coef);
}

// ---------------------------------------------------------------------------
// out = (relu?)(agg + xw * dinv^2 + bias)   elementwise over [N, F]
// ---------------------------------------------------------------------------
template <int F, bool RELU>
__global__ void combine_kernel(const float* __restrict__ agg,
                               const float* __restrict__ xw,
                               const float* __restrict__ dinv,
                               const float* __restrict__ bias,
                               float* __restrict__ out, size_t total) {
    constexpr int SHIFT = (F == 64) ? 6 : 5;
    size_t g = (size_t)blockIdx.x * blockDim.x + threadIdx.x;
    if (g >= total) return;
    size_t node = g >> SHIFT;
    int    f    = (int)(g & (F - 1));
    float di = dinv[node];
    float v  = agg[g] + xw[g] * (di * di) + bias[f];
    out[g] = RELU ? fmaxf(v, 0.0f) : v;
}

// ---------------------------------------------------------------------------
// Launch: two GCN layers.
// ws layout (floats): dinv[N_pad] | agg[N*64] | xw[N*64]
// ---------------------------------------------------------------------------
extern "C" void kernel_launch(void* const* d_in, const int* in_sizes, int n_in,
                              void* d_out, int out_size, void* d_ws, size_t ws_size,
                              hipStream_t stream) {
    (void)n_in; (void)out_size; (void)ws_size;

    const float* x  = (const float*)d_in[0];
    const int*   ei = (const int*)d_in[1];
    const float* W1 = (const float*)d_in[2];
    const float* b1 = (const float*)d_in[3];
    const float* W2 = (const float*)d_in[4];
    const float* b2 = (const float*)d_in[5];
    float* out = (float*)d_out;

    const int Fin = 64, Fh = 64, Fo = 32;
    const int N = in_sizes[0] / Fin;          // 100000
    const int E = in_sizes[1] / 2;            // 1600000
    const int* src = ei;
    const int* dst = ei + E;

    const size_t N_pad = ((size_t)N + 63) & ~(size_t)63;
    float* dinv = (float*)d_ws;
    float* agg  = dinv + N_pad;               // [N, 64], reused as h
    float* xw   = agg + (size_t)N * Fh;       // [N, 64], reused for xw2

    const int T = 256;
    auto blocks = [](size_t n, int t) { return (unsigned)((n + t - 1) / t); };

    // 1) zero dinv+agg region and d_out (agg2 accumulates into d_out)
    size_t zws = N_pad + (size_t)N * Fh;
    zero_kernel<<<blocks(zws, T), T, 0, stream>>>((float*)d_ws, zws);
    zero_kernel<<<blocks((size_t)N * Fo, T), T, 0, stream>>>(out, (size_t)N * Fo);

    // 2) degree on dst, then dinv = rsqrt(deg + 1)
    deg_kernel<<<blocks((size_t)E, T), T, 0, stream>>>(dst, (unsigned*)dinv, E);
    dinv_kernel<<<blocks((size_t)N, T), T, 0, stream>>>(dinv, N);

    // 3) layer 1: xw1 = x @ W1
    {
        unsigned waves = (unsigned)(N / 16) * (Fh / 16);
        gemm_wmma_kernel<64><<<blocks(waves, T / 32), T, 0, stream>>>(x, W1, xw, N);
    }
    // 4) scatter edges into agg
    scatter_kernel<64><<<blocks((size_t)E, T / 32), T, 0, stream>>>(xw, src, dst, dinv, agg, E);
    // 5) h = relu(agg + xw1*dinv^2 + b1), in place over agg
    combine_kernel<64, true><<<blocks((size_t)N * Fh, T), T, 0, stream>>>(
        agg, xw, dinv, b1, agg, (size_t)N * Fh);

    // 6) layer 2: xw2 = h @ W2
    {
        unsigned waves = (unsigned)(N / 16) * (Fo / 16);
        gemm_wmma_kernel<32><<<blocks(waves, T / 32), T, 0, stream>>>(agg, W2, xw, N);
    }
    // 7) scatter edges into d_out
    scatter_kernel<32><<<blocks((size_t)E, T / 32), T, 0, stream>>>(xw, src, dst, dinv, out, E);
    // 8) out = agg2 + xw2*dinv^2 + b2, in place
    combine_kernel<32, false><<<blocks((size_t)N * Fo, T), T, 0, stream>>>(
        out, xw, dinv, b2, out, (size_t)N * Fo);
}